// InverseHaarUpsample_37108517437889
// MI455X (gfx1250) — compile-verified
//
#include <hip/hip_runtime.h>

typedef __attribute__((ext_vector_type(2))) float v2f;
typedef __attribute__((ext_vector_type(8))) float v8f;

namespace {
constexpr int  B_  = 16;
constexpr int  C_  = 64;                       // output channels (input has 4*C_)
constexpr int  H_  = 128;
constexpr int  W_  = 128;
constexpr long HW_ = (long)H_ * W_;            // 16384
constexpr long SUBBAND_STRIDE = (long)C_ * HW_;        // 1,048,576 elements (4 MB)
constexpr long IN_BATCH       = 4L * C_ * HW_;         // elements per input batch
constexpr int  OW_            = 2 * W_;                // 256
constexpr long OUT_PLANE      = (long)(2 * H_) * OW_;  // 65536
constexpr long OUT_BATCH      = (long)C_ * OUT_PLANE;
constexpr int  GROUPS         = W_ / 16;               // 8 groups of 16 pixels per row
constexpr int  ROWS_PER_WAVE  = 2;                     // 16 WMMAs per wave, immediates only
}

// One wave handles ROWS_PER_WAVE full input rows of one (b,c) plane.
// Per 16-pixel group: 2 coalesced b32 loads (LL/LH on lanes 0-15, HL/HH on
// lanes 16-31), one v_wmma_f32_16x16x4_f32 with the constant Haar matrix in A,
// one b64 NT store. A-rows are permuted so lanes 0-15 read combos 0,1 out of
// d[0],d[1] (output row 2h) and lanes 16-31 read combos 2,3 (row 2h+1) out of
// the same VGPR slots -> no per-lane selects in the hot loop.
__global__ __launch_bounds__(256) void inv_haar_wmma(const float* __restrict__ in,
                                                     float* __restrict__ out) {
  const int tid  = blockIdx.x * blockDim.x + threadIdx.x;
  const int lane = threadIdx.x & 31;
  int g = tid >> 5;                                   // global wave id

  const int hb = g % (H_ / ROWS_PER_WAVE);  g /= (H_ / ROWS_PER_WAVE);
  const int c  = g % C_;                    g /= C_;
  const int b  = g;
  const int h0 = hb * ROWS_PER_WAVE;

  const int m  = lane & 15;                 // A-matrix row / B-matrix column
  const int hi = lane >> 4;                 // 0: K=0,1 (LL,LH) ; 1: K=2,3 (HL,HH)

  // D rows consumed: lanes 0-15 read D[0],D[1]; lanes 16-31 read D[8],D[9].
  // Map A row m -> combo i so those slots carry combos (0,1 | 2,3):
  const int i = (m & 1) | ((m & 8) >> 2);

  // Haar synthesis coefficients: combo i = 2p+q,
  //   K0 (LL): +;  K1 (LH): + iff p==1;  K2 (HL): + iff q==1;  K3 (HH): + iff p==q
  const float a0 = hi ? ((i & 1) ? 0.5f : -0.5f) : 0.5f;
  const float a1 = hi ? (((i == 0) || (i == 3)) ? 0.5f : -0.5f)
                      : ((i & 2) ? 0.5f : -0.5f);
  v2f a; a[0] = a0; a[1] = a1;

  // Base pointers (all later displacements are compile-time immediates)
  const float* pin  = in  + (long)b * IN_BATCH + (long)c * HW_
                          + (long)h0 * W_ + (long)(hi * 2) * SUBBAND_STRIDE + m;
  float*       pout = out + (long)b * OUT_BATCH + (long)c * OUT_PLANE
                          + (long)(2 * h0 + hi) * OW_ + 2 * m;

#pragma unroll
  for (int r = 0; r < ROWS_PER_WAVE; ++r) {
#pragma unroll
    for (int gi = 0; gi < GROUPS; ++gi) {
      const long iofs = (long)r * W_ + gi * 16;       // input elems (immediate)
      v2f bm;
      bm[0] = __builtin_nontemporal_load(pin + iofs);                   // LL | HL
      bm[1] = __builtin_nontemporal_load(pin + iofs + SUBBAND_STRIDE);  // LH | HH

      v8f cacc = {};
      v8f d = __builtin_amdgcn_wmma_f32_16x16x4_f32(false, a, false, bm, (short)0,
                                                    cacc, false, false);

      v2f val; val[0] = d[0]; val[1] = d[1];
      v2f* dst = reinterpret_cast<v2f*>(pout + (long)r * 2 * OW_ + gi * 32);
      __builtin_nontemporal_store(val, dst);          // 128B contiguous per half-wave
    }
  }
}

extern "C" void kernel_launch(void* const* d_in, const int* in_sizes, int n_in,
                              void* d_out, int out_size, void* d_ws, size_t ws_size,
                              hipStream_t stream) {
  const float* x   = (const float*)d_in[0];
  float*       out = (float*)d_out;

  const long totalWaves   = (long)B_ * C_ * (H_ / ROWS_PER_WAVE);  // 65,536
  const long totalThreads = totalWaves * 32;                        // 2,097,152
  const int  block        = 256;                                    // 8 waves/block
  const int  grid         = (int)(totalThreads / block);            // 8,192 blocks

  inv_haar_wmma<<<grid, block, 0, stream>>>(x, out);
}